// LocalLayer_44942537785491
// MI455X (gfx1250) — compile-verified
//
#include <hip/hip_runtime.h>

typedef __attribute__((ext_vector_type(2))) float v2f;
typedef __attribute__((ext_vector_type(8))) float v8f;

#define DIM 128            // D == D_OUT == 128
#define K3  384            // 3*D concat width
#define LDS_STRIDE 388     // 384 + 4 pad floats; 388*4 B row stride is 16B-aligned and
                           // 388 % 64 != 0 so the 16-row A-fragment reads hit distinct banks

// ---------------- workspace zeroing (accumulators must start at 0 every call) ----------
__global__ __launch_bounds__(256) void zero_ws_kernel(float4* __restrict__ p, int n4) {
    int i = blockIdx.x * blockDim.x + threadIdx.x;
    if (i < n4) p[i] = make_float4(0.f, 0.f, 0.f, 0.f);
}

// ---------------- edge scatter-add: one wave32 per edge, 4 floats per lane --------------
// acc[] (51.2 MB per edge set) + x (51.2 MB) fit in the 192 MB L2, so these
// global_atomic_add_f32 ops resolve at L2 rate after first touch.
__global__ __launch_bounds__(256) void edge_agg_kernel(const float* __restrict__ x,
                                                       const int* __restrict__ src,
                                                       const int* __restrict__ dst,
                                                       float* __restrict__ acc,
                                                       int n_edges) {
    int wave = blockIdx.x * 8 + (threadIdx.x >> 5);
    if (wave >= n_edges) return;
    int lane = threadIdx.x & 31;
    int s = src[wave];
    int d = dst[wave];
    const float4 v = *reinterpret_cast<const float4*>(x + (size_t)s * DIM + lane * 4);
    float* dp = acc + (size_t)d * DIM + lane * 4;
    unsafeAtomicAdd(dp + 0, v.x);
    unsafeAtomicAdd(dp + 1, v.y);
    unsafeAtomicAdd(dp + 2, v.z);
    unsafeAtomicAdd(dp + 3, v.w);
}

// ---------------- fused concat + GEMM + bias via V_WMMA_F32_16X16X4_F32 -----------------
// Block = 256 threads = 8 wave32s. Block owns a 16-row output strip; wave w owns
// output columns [16w, 16w+16). The 16x384 strip of [x | x_pos | x_neg] is staged
// in LDS once with b128 transfers; W (192 KB, read by all blocks) lives in L2.
__global__ __launch_bounds__(256) void fused_gemm_kernel(const float* __restrict__ x,
                                                         const float* __restrict__ xp,
                                                         const float* __restrict__ xn,
                                                         const float* __restrict__ W,
                                                         const float* __restrict__ bias,
                                                         float* __restrict__ out,
                                                         int n_nodes) {
    __shared__ float lds[16 * LDS_STRIDE];
    const int r0 = blockIdx.x * 16;

    // cooperative float4-granular load of the concatenated 16x384 A-strip.
    // 16*96 = 1536 float4s, 6 per thread; source boundaries (col 128, 256) are
    // float4-aligned so each float4 comes from exactly one source array.
    for (int idx = threadIdx.x; idx < 16 * (K3 / 4); idx += 256) {
        int row = idx / (K3 / 4);
        int c4  = idx - row * (K3 / 4);
        int col = c4 * 4;
        int gr  = r0 + row;
        if (gr >= n_nodes) gr = n_nodes - 1;   // clamp; OOB rows never stored
        float4 v;
        if (col < DIM)          v = *(const float4*)(x  + (size_t)gr * DIM + col);
        else if (col < 2 * DIM) v = *(const float4*)(xp + (size_t)gr * DIM + (col - DIM));
        else                    v = *(const float4*)(xn + (size_t)gr * DIM + (col - 2 * DIM));
        *(float4*)(&lds[row * LDS_STRIDE + col]) = v;
    }
    __syncthreads();

    const int lane  = threadIdx.x & 31;
    const int wv    = threadIdx.x >> 5;
    const int n0    = wv * 16;           // this wave's output-column base
    const int mn    = lane & 15;         // A row (M) == B/C column (N) index for this lane
    const int khalf = (lane >> 4) << 1;  // lanes 0-15 -> K pair {0,1}; lanes 16-31 -> {2,3}

    const float* lp = &lds[mn * LDS_STRIDE + khalf];
    const float* Wp = W + (size_t)khalf * DIM + n0 + mn;

    v8f c = {0.f, 0.f, 0.f, 0.f, 0.f, 0.f, 0.f, 0.f};
#pragma unroll 8
    for (int k = 0; k < K3; k += 4) {
        v2f a, bb;
        // A fragment (16x4 fp32): lane holds A[M=mn][k+khalf], A[M=mn][k+khalf+1]
        a.x = lp[k];
        a.y = lp[k + 1];
        // B fragment (4x16 fp32): lane holds W[k+khalf][n0+mn], W[k+khalf+1][n0+mn]
        bb.x = Wp[(size_t)k * DIM];
        bb.y = Wp[(size_t)k * DIM + DIM];
        // D = A*B + C   (8 args: neg_a, A, neg_b, B, c_mod, C, reuse_a, reuse_b)
        c = __builtin_amdgcn_wmma_f32_16x16x4_f32(false, a, false, bb,
                                                  (short)0, c, false, false);
    }

    // C/D layout: lanes 0-15 hold M=i in VGPR i; lanes 16-31 hold M=8+i
    const float bv    = bias[n0 + mn];
    const int   rbase = (lane >> 4) * 8;
    float* op = out + (size_t)(r0 + rbase) * DIM + n0 + mn;

    if (r0 + 16 <= n_nodes) {
        // uniform fast path: full tile (always taken when 16 | n_nodes)
#pragma unroll
        for (int i = 0; i < 8; ++i)
            op[(size_t)i * DIM] = c[i] + bv;
    } else {
        for (int i = 0; i < 8; ++i)
            if (r0 + rbase + i < n_nodes)
                op[(size_t)i * DIM] = c[i] + bv;
    }
}

extern "C" void kernel_launch(void* const* d_in, const int* in_sizes, int n_in,
                              void* d_out, int out_size, void* d_ws, size_t ws_size,
                              hipStream_t stream) {
    const float* x   = (const float*)d_in[0];
    const int*   pos = (const int*)d_in[1];   // [2, E] row-major: row0=src, row1=dst
    const int*   neg = (const int*)d_in[2];
    const float* W   = (const float*)d_in[3]; // [384, 128] row-major
    const float* b   = (const float*)d_in[4]; // [128]
    float* out = (float*)d_out;

    const int N = in_sizes[0] / DIM;  // 100000
    const int E = in_sizes[1] / 2;    // 500000

    float* xp = (float*)d_ws;                 // [N, 128] pos accumulator
    float* xn = xp + (size_t)N * DIM;         // [N, 128] neg accumulator

    const int n4 = (2 * N * DIM) / 4;
    zero_ws_kernel<<<(n4 + 255) / 256, 256, 0, stream>>>((float4*)d_ws, n4);

    const int agg_blocks = (E + 7) / 8;       // 8 waves (edges) per 256-thread block
    edge_agg_kernel<<<agg_blocks, 256, 0, stream>>>(x, pos, pos + E, xp, E);
    edge_agg_kernel<<<agg_blocks, 256, 0, stream>>>(x, neg, neg + E, xn, E);

    fused_gemm_kernel<<<(N + 15) / 16, 256, 0, stream>>>(x, xp, xn, W, b, out, N);
}